// HierarchicalResidualManager_53712861004565
// MI455X (gfx1250) — compile-verified
//
#include <hip/hip_runtime.h>
#include <stdint.h>

// Problem geometry (from reference)
#define DIMW    8192
#define BATCH   128
#define LEVELS  128
#define NELEM   (BATCH * DIMW)      // 1,048,576 elements per level
#define VEC     4                   // elements per thread (float4)
#define BLOCK   256                 // 8 wave32s per block
#define TILE_ELEMS (BLOCK * VEC)    // 1024 floats = 4 KB LDS tile per block

typedef __attribute__((ext_vector_type(4))) float        v4f;
typedef __attribute__((ext_vector_type(4))) unsigned int v4u;
typedef __attribute__((ext_vector_type(4))) int          v4i;
typedef __attribute__((ext_vector_type(8))) int          v8i;

// Psi(x) = sin(K1 x) * exp(-K2 x^2) + cos(K3 x)
__device__ __forceinline__ float qrt_damp(float r) {
    const float K1 = 118.645535f;    // phi*sqrt(2)*51.85
    const float K2 = 0.61803399f;    // 1/phi
    const float K3 = 1.94161104f;    // pi/phi
    float s = __sinf(K1 * r);                 // v_sin_f32
    float e = __expf(-(r * r) * K2);          // v_exp_f32
    float c = __cosf(K3 * r);                 // v_cos_f32
    return fmaf(s, e, c);
}

__global__ __launch_bounds__(BLOCK)
void hier_residual_kernel(const float* __restrict__ tokens,
                          float* __restrict__ out)
{
    __shared__ __align__(16) float tile[TILE_ELEMS];

    const uint32_t tid        = threadIdx.x;
    const uint64_t block_elem = (uint64_t)blockIdx.x * TILE_ELEMS;

#if defined(__gfx1250__) && __has_builtin(__builtin_amdgcn_tensor_load_to_lds) && \
    __has_builtin(__builtin_amdgcn_s_wait_tensorcnt)
    // ---- Stage this block's token tile into LDS via the Tensor Data Mover ----
    // 1-D tensor: tensor_dim0 = tile_dim0 = 1024 elements, data_size = 4 bytes.
    if (tid < 32) {  // one wave issues the TDM op (EXEC-independent, SGPR descriptors)
        const uint64_t ga      = (uint64_t)(uintptr_t)(tokens + block_elem);
        const uint32_t lds_off = (uint32_t)(uintptr_t)&tile[0];  // flat LDS addr low 32b = LDS offset
        v4u g0;
        g0.x = 1u;                                             // count=1, user descriptor
        g0.y = lds_off;                                        // lds_addr
        g0.z = (uint32_t)(ga & 0xFFFFFFFFu);                   // global_addr[31:0]
        g0.w = (uint32_t)((ga >> 32) & 0x1FFFFFFu)             // global_addr[56:32]
             | (2u << 30);                                     // type = 2 ("image")
        v8i g1;
        g1[0] = (int)(2u << 16);                               // data_size=4B; no multicast
        g1[1] = (int)(((uint32_t)TILE_ELEMS & 0xFFFFu) << 16); // tensor_dim0[15:0] @ bits 63:48
        g1[2] = (int)(((uint32_t)TILE_ELEMS >> 16) | (1u << 16)); // tensor_dim0[31:16]; tensor_dim1=1
        g1[3] = (int)((uint32_t)TILE_ELEMS << 16);             // tile_dim0 @ bits 127:112
        g1[4] = 1;                                             // tile_dim1 = 1
        g1[5] = TILE_ELEMS;                                    // tensor_dim0_stride[31:0]
        g1[6] = 0;
        g1[7] = 0;
        v4i zz4 = {0, 0, 0, 0};                                // groups 2/3 unused (<=2D tensor)
        v8i zz8 = {0, 0, 0, 0, 0, 0, 0, 0};                    // extra group (clang-23 6-arg form)
        __builtin_amdgcn_tensor_load_to_lds(g0, g1, zz4, zz4, zz8, 0);
        __builtin_amdgcn_s_wait_tensorcnt(0);                  // s_wait_tensorcnt 0
    }
    __syncthreads();
    const v4f v = *reinterpret_cast<const v4f*>(&tile[tid * VEC]);  // ds_load_b128
#else
    // Fallback path (toolchain without the TDM builtin): direct coalesced load.
    const v4f v = *reinterpret_cast<const v4f*>(tokens + block_elem + (uint64_t)tid * VEC);
    (void)tile;
#endif

    // Four independent serial chains per thread -> hides v_sin/v_cos/v_exp latency.
    float p0 = 0.f, p1 = 0.f, p2 = 0.f, p3 = 0.f;
    float d = 1.0f;
    const float DECAYF = 0.38196601f;  // 1/phi^2 (fp32 iterative multiply, underflows to 0 ~level 108)

    float* outp = out + block_elem + (uint64_t)tid * VEC;

    #pragma unroll 4
    for (int l = 0; l < LEVELS; ++l) {
        float r0 = (v.x - p0) * d;
        float r1 = (v.y - p1) * d;
        float r2 = (v.z - p2) * d;
        float r3 = (v.w - p3) * d;

        float w0 = qrt_damp(r0);
        float w1 = qrt_damp(r1);
        float w2 = qrt_damp(r2);
        float w3 = qrt_damp(r3);

        p0 += w0; p1 += w1; p2 += w2; p3 += w3;

        v4f wv = {w0, w1, w2, w3};
        // Write-once 512 MiB stream: non-temporal b128 store, don't thrash L2.
        __builtin_nontemporal_store(wv, reinterpret_cast<v4f*>(outp));

        outp += NELEM;      // next level plane
        d *= DECAYF;
    }
}

extern "C" void kernel_launch(void* const* d_in, const int* in_sizes, int n_in,
                              void* d_out, int out_size, void* d_ws, size_t ws_size,
                              hipStream_t stream) {
    const float* tokens = (const float*)d_in[0];   // [128, 8192] fp32
    float*       out    = (float*)d_out;           // [128, 128, 8192] fp32

    const int nblocks = NELEM / TILE_ELEMS;        // 1024 blocks of 256 threads
    hipLaunchKernelGGL(hier_residual_kernel, dim3(nblocks), dim3(BLOCK), 0, stream,
                       tokens, out);
}